// ECELoss_9466107920861
// MI455X (gfx1250) — compile-verified
//
#include <hip/hip_runtime.h>
#include <hip/hip_bf16.h>

// ECE loss, MI455X (gfx1250, wave32).  FINAL.
//
// Roofline: 512MB logits + 4MB labels streamed once @ 23.3 TB/s => ~22us floor;
// ~4e8 flops (incl 1.28e8 v_exp_f32 trans ops ~ 6us chip-wide) fully hidden.
// => HBM-bandwidth-bound; WMMA inapplicable (0.8 flop/byte, reductions).
//
// Design: one row (128 f32) per wave32; lane l owns the float4 at col 4l so
// each row is one coalesced 512B burst. Double-buffered
// GLOBAL_LOAD_ASYNC_TO_LDS_B128 (th:TH_LOAD_NT -- one-touch stream > 192MB L2)
// lets the async DMA engine prefetch the wave's next row into LDS while the
// current row is consumed via ds_load_b128; s_wait_asynccnt 1 keeps exactly
// one ASYNCcnt-tracked copy in flight (async loads complete in order).
// (max,argmax) and sum(exp(x-max)) via wave32 butterfly shuffles; conf=1/sum.
// 15-bin accumulation: LDS f32 atomics per block -> one f64 global atomic per
// bin-slot per block -> 1-wave finalize.

#define NBINS 15
#define NACC  (3 * NBINS)   // counts[15], conf_sum[15], acc_sum[15]

typedef float v4f __attribute__((ext_vector_type(4)));

// ------------------------------------------------- async-to-LDS asm helpers
// VGLOBAL encoding: VDST = LDS byte-address VGPR, VADDR = 64-bit global addr
// (GV mode, SADDR=off). One instruction per wave => one ASYNCcnt unit.
// th:TH_LOAD_NT verified to assemble (round-3 probe).
__device__ __forceinline__ void async_load_b128_nt(unsigned lds_addr, const void* gaddr) {
  asm volatile("global_load_async_to_lds_b128 %0, %1, off th:TH_LOAD_NT"
               :: "v"(lds_addr), "v"(gaddr) : "memory");
}
__device__ __forceinline__ void wait_asynccnt_le1() {
  asm volatile("s_wait_asynccnt 1" ::: "memory");
}
__device__ __forceinline__ void wait_asynccnt_0() {
  asm volatile("s_wait_asynccnt 0" ::: "memory");
}

// ------------------------------------------------- shared row-processing body
__device__ __forceinline__ void process_row(v4f v, float inv_t, long long row,
                                            const int* __restrict__ labels,
                                            float* sb, int lane) {
  v *= inv_t;
  // Local max/argmax over the lane's 4 elements (first-occurrence tiebreak).
  float m = v.x; int mi = lane * 4;
  if (v.y > m) { m = v.y; mi = lane * 4 + 1; }
  if (v.z > m) { m = v.z; mi = lane * 4 + 2; }
  if (v.w > m) { m = v.w; mi = lane * 4 + 3; }
  // Wave32 butterfly reduce on (value, smallest-index-on-tie): associative &
  // commutative => all 32 lanes converge to the row max / first argmax.
  #pragma unroll
  for (int off = 16; off; off >>= 1) {
    float om = __shfl_xor(m,  off, 32);
    int   oi = __shfl_xor(mi, off, 32);
    if (om > m || (om == m && oi < mi)) { m = om; mi = oi; }
  }
  // Stable softmax denominator; conf = exp(max-max)/sum = 1/sum.
  float s = __expf(v.x - m) + __expf(v.y - m) + __expf(v.z - m) + __expf(v.w - m);
  #pragma unroll
  for (int off = 16; off; off >>= 1)
    s += __shfl_xor(s, off, 32);
  if (lane == 0) {
    float conf = 1.0f / s;
    int bin = (int)ceilf(conf * (float)NBINS) - 1;   // bin k: (k/15, (k+1)/15]
    bin = bin < 0 ? 0 : (bin > NBINS - 1 ? NBINS - 1 : bin);
    float acc = (labels[row] == mi) ? 1.0f : 0.0f;
    atomicAdd(&sb[bin],             1.0f);
    atomicAdd(&sb[NBINS + bin],     conf);
    atomicAdd(&sb[2 * NBINS + bin], acc);
  }
}

// ---------------------------------------------------------------- zero scratch
__global__ void ece_zero_ws(double* __restrict__ ws) {
  int i = threadIdx.x;
  if (i < NACC) ws[i] = 0.0;
}

// -------------------------------------- async-staged main stream (LAUNCHED)
__global__ __launch_bounds__(256) void ece_main_async(
    const float* __restrict__ logits,   // [n_rows * 128]
    const int*   __restrict__ labels,   // [n_rows]
    const int*   __restrict__ t_opt,    // [1]
    double*      __restrict__ ws,       // [NACC] zeroed
    int n_rows) {
  __shared__ float sb[NACC];
  __shared__ v4f stage[2][256];          // 2 bufs x 8 waves x 32 lanes x 16B = 8KB
  if (threadIdx.x < NACC) sb[threadIdx.x] = 0.0f;
  __syncthreads();

  const int lane          = threadIdx.x & 31;
  const int wave_in_blk   = threadIdx.x >> 5;
  const int waves_per_blk = blockDim.x >> 5;
  const long long wave_id     = (long long)blockIdx.x * waves_per_blk + wave_in_blk;
  const long long wave_stride = (long long)gridDim.x * waves_per_blk;

  const int   t     = *t_opt;                  // static scalar; 0 => no scaling
  const float inv_t = (t != 0) ? (1.0f / (float)t) : 1.0f;

  // Workgroup-relative LDS byte offsets of this thread's two staging slots
  // (low 32 bits of the generic pointer to LDS; HW adds LDS_BASE).
  const unsigned slot0 = (unsigned)(uintptr_t)&stage[0][threadIdx.x];
  const unsigned slot1 = (unsigned)(uintptr_t)&stage[1][threadIdx.x];

  long long row = wave_id;
  int p = 0;
  if (row < n_rows)
    async_load_b128_nt(slot0, logits + row * 128 + lane * 4);  // prologue copy

  for (; row < n_rows; row += wave_stride) {
    const long long nxt = row + wave_stride;
    const bool have_next = (nxt < (long long)n_rows);
    if (have_next)
      async_load_b128_nt(p ? slot0 : slot1, logits + nxt * 128 + lane * 4);
    if (have_next) wait_asynccnt_le1();   // oldest (current row) has landed
    else           wait_asynccnt_0();
    v4f v = stage[p][threadIdx.x];        // ds_load_b128 from staged row
    process_row(v, inv_t, row, labels, sb, lane);
    p ^= 1;
  }

  __syncthreads();
  // One f64 global atomic per bin-slot per block (global_atomic_add_f64).
  if (threadIdx.x < NACC) {
    float q = sb[threadIdx.x];
    if (q != 0.0f) atomicAdd(&ws[threadIdx.x], (double)q);
  }
}

// ---------------------------------------------------------------- finalize
__global__ void ece_final(const double* __restrict__ ws,
                          float* __restrict__ out, int n) {
  int i = threadIdx.x;               // one wave32; lanes 0..14 own a bin
  float term = 0.0f;
  if (i < NBINS) {
    double cnt = ws[i];
    double cs  = ws[NBINS + i];
    double as  = ws[2 * NBINS + i];
    if (cnt > 0.0) {
      double avg_c = cs / cnt;
      double avg_a = as / cnt;
      term = (float)(fabs(avg_c - avg_a) * (cnt / (double)n));
    }
  }
  #pragma unroll
  for (int off = 16; off; off >>= 1)
    term += __shfl_xor(term, off, 32);
  if (i == 0) out[0] = term;
}

// ---------------------------------------------------------------- launcher
extern "C" void kernel_launch(void* const* d_in, const int* in_sizes, int n_in,
                              void* d_out, int out_size, void* d_ws, size_t ws_size,
                              hipStream_t stream) {
  const float* logits = (const float*)d_in[0];
  const int*   labels = (const int*)d_in[1];
  const int*   t_opt  = (const int*)d_in[2];
  double*      ws     = (double*)d_ws;
  float*       out    = (float*)d_out;

  const int n_rows = in_sizes[1];          // N = 1,000,000 (labels count)

  ece_zero_ws<<<1, 64, 0, stream>>>(ws);
  ece_main_async<<<4096, 256, 0, stream>>>(logits, labels, t_opt, ws, n_rows);
  ece_final<<<1, 32, 0, stream>>>(ws, out, n_rows);
}